// Aggregator_9964324127508
// MI455X (gfx1250) — compile-verified
//
#include <hip/hip_runtime.h>
#include <hip/hip_bf16.h>

typedef float v2f __attribute__((ext_vector_type(2)));
typedef float v8f __attribute__((ext_vector_type(8)));

#define RW       2597      // region rows (R1 - R0)
#define R0_IDX   42033
#define CH       64
#define N_ENT    100000
#define N_USR    50000

// ---------------- utility kernels ----------------

__global__ __launch_bounds__(256) void zero_kernel(float* __restrict__ p, int n) {
  int i = blockIdx.x * 256 + threadIdx.x;
  if (i < n) p[i] = 0.0f;
}

// ent = 0.8*emb + 0.2*emb for all rows (region rows overwritten by GEMM kernel next)
__global__ __launch_bounds__(256) void build_ent_kernel(const float4* __restrict__ emb,
                                                        float4* __restrict__ ent, int n4) {
  int i = blockIdx.x * 256 + threadIdx.x;
  if (i < n4) {
    float4 e = emb[i];
    float4 o;
    o.x = 0.8f * e.x + 0.2f * e.x;
    o.y = 0.8f * e.y + 0.2f * e.y;
    o.z = 0.8f * e.z + 0.2f * e.z;
    o.w = 0.8f * e.w + 0.2f * e.w;
    ent[i] = o;
  }
}

// ---------------- region GEMM via V_WMMA_F32_16X16X4_F32 ----------------
// 8 waves per block, one 16-row M tile per wave; each wave computes all 4 N tiles.
// A: rwm[m, k] 16x4 per step.  B: emb[(R0+k), n] 4x16 per step (row-major, stride CH).
// Out-of-range A rows are clamped to row 0: their products land only in D rows >= RW,
// which are never stored, so no zero-masking (and no exec divergence) is needed.
__global__ __launch_bounds__(256) void region_gemm_kernel(const float* __restrict__ rwm,
                                                          const float* __restrict__ emb,
                                                          float* __restrict__ ent) {
  const int lane = threadIdx.x & 31;   // wave32, EXEC all ones within active waves
  const int wid  = threadIdx.x >> 5;
  const int tile = blockIdx.x * 8 + wid;
  const int m0   = tile * 16;
  if (m0 >= RW) return;                // whole-wave exit keeps EXEC uniform

  const int lm   = lane & 15;          // A: M row within tile; B/D: N column
  const int kh   = lane >> 4;          // K half-select (0 -> K0/K1, 1 -> K2/K3)
  const int arow = m0 + lm;
  const int arc  = (arow < RW) ? arow : 0;   // clamped A row (garbage only hits unstored D rows)
  const float* __restrict__ Arow = rwm + (size_t)arc * RW + 2 * kh;
  const float* __restrict__ Breg = emb + (size_t)R0_IDX * CH;

  v8f acc0 = {0,0,0,0,0,0,0,0};
  v8f acc1 = acc0, acc2 = acc0, acc3 = acc0;

  int k = 0;
  for (; k + 4 <= RW; k += 4) {
    v2f a;
    a.x = Arow[k];                         // A[arc, k + 2*kh]     (unconditional)
    a.y = Arow[k + 1];                     // A[arc, k + 2*kh + 1]
    const float* __restrict__ b0 = Breg + (size_t)(k + 2 * kh) * CH + lm;
    const float* __restrict__ b1 = b0 + CH;
    v2f b;
    b.x = b0[0];  b.y = b1[0];
    acc0 = __builtin_amdgcn_wmma_f32_16x16x4_f32(false, a, false, b, (short)0, acc0, false, false);
    b.x = b0[16]; b.y = b1[16];
    acc1 = __builtin_amdgcn_wmma_f32_16x16x4_f32(false, a, false, b, (short)0, acc1, false, false);
    b.x = b0[32]; b.y = b1[32];
    acc2 = __builtin_amdgcn_wmma_f32_16x16x4_f32(false, a, false, b, (short)0, acc2, false, false);
    b.x = b0[48]; b.y = b1[48];
    acc3 = __builtin_amdgcn_wmma_f32_16x16x4_f32(false, a, false, b, (short)0, acc3, false, false);
  }

  // K tail (RW % 4 == 1): clamped in-bounds loads + value selects (no branches),
  // straight-line WMMA so the compiler keeps a single accumulator set.
  {
    const int ka  = k + 2 * kh;                         // 2596 (kh=0) / 2598 (kh=1)
    const int kc0 = (ka     < RW) ? ka     : RW - 1;
    const int kc1 = (ka + 1 < RW) ? ka + 1 : RW - 1;
    const float* __restrict__ Ar = rwm + (size_t)arc * RW;
    float ax = Ar[kc0];
    float ay = Ar[kc1];
    v2f a;
    a.x = (ka     < RW) ? ax : 0.0f;   // zero A beyond K; B garbage then multiplies by 0
    a.y = (ka + 1 < RW) ? ay : 0.0f;
    const float* __restrict__ b0 = Breg + (size_t)kc0 * CH + lm;
    const float* __restrict__ b1 = Breg + (size_t)kc1 * CH + lm;
    v2f b;
    b.x = b0[0];  b.y = b1[0];
    acc0 = __builtin_amdgcn_wmma_f32_16x16x4_f32(false, a, false, b, (short)0, acc0, false, false);
    b.x = b0[16]; b.y = b1[16];
    acc1 = __builtin_amdgcn_wmma_f32_16x16x4_f32(false, a, false, b, (short)0, acc1, false, false);
    b.x = b0[32]; b.y = b1[32];
    acc2 = __builtin_amdgcn_wmma_f32_16x16x4_f32(false, a, false, b, (short)0, acc2, false, false);
    b.x = b0[48]; b.y = b1[48];
    acc3 = __builtin_amdgcn_wmma_f32_16x16x4_f32(false, a, false, b, (short)0, acc3, false, false);
  }

  // D layout: VGPR r, lanes 0-15 -> M = r, lanes 16-31 -> M = r + 8; N = lm (+16*t)
  #pragma unroll
  for (int r = 0; r < 8; ++r) {
    int row = m0 + r + 8 * kh;
    if (row < RW) {
      size_t base = (size_t)(R0_IDX + row) * CH + lm;
      ent[base +  0] = 0.8f * emb[base +  0] + 0.2f * acc0[r];
      ent[base + 16] = 0.8f * emb[base + 16] + 0.2f * acc1[r];
      ent[base + 32] = 0.8f * emb[base + 32] + 0.2f * acc2[r];
      ent[base + 48] = 0.8f * emb[base + 48] + 0.2f * acc3[r];
    }
  }
}

// ---------------- KG aggregate: one wave per edge ----------------
__global__ __launch_bounds__(256) void kg_agg_kernel(const int* __restrict__ head,
    const int* __restrict__ tail, const int* __restrict__ etype,
    const float* __restrict__ ent, const float* __restrict__ relw,
    float* __restrict__ sums, float* __restrict__ counts, int nedges) {
  int e = blockIdx.x * 8 + (threadIdx.x >> 5);
  if (e >= nedges) return;
  int lane = threadIdx.x & 31;
  int h = head[e];
  int t = tail[e];
  int r = (etype[e] + 31) & 31;   // (etype - 1) mod 32, etype >= 0
  float2 ev = ((const float2*)(ent  + (size_t)t * CH))[lane];
  float2 wv = ((const float2*)(relw + (size_t)r * CH))[lane];
  float* o = sums + (size_t)h * CH + 2 * lane;
  unsafeAtomicAdd(o,     ev.x * wv.x);
  unsafeAtomicAdd(o + 1, ev.y * wv.y);
  if (lane == 0) unsafeAtomicAdd(counts + h, 1.0f);
}

__global__ __launch_bounds__(256) void finalize_kernel(float* __restrict__ sums,
    const float* __restrict__ counts, int n) {
  int i = blockIdx.x * 256 + threadIdx.x;
  if (i < n) {
    float c = counts[i >> 6];
    sums[i] = sums[i] / fmaxf(c, 1.0f);
  }
}

// ---------------- user aggregate: one wave per nnz ----------------
__global__ __launch_bounds__(256) void user_agg_kernel(const int* __restrict__ rows,
    const int* __restrict__ cols, const float* __restrict__ vals,
    const float* __restrict__ ent, float* __restrict__ out, int nnz) {
  int i = blockIdx.x * 8 + (threadIdx.x >> 5);
  if (i >= nnz) return;
  int lane = threadIdx.x & 31;
  int rr = rows[i];
  int cc = cols[i];
  float v = vals[i];
  float2 ev = ((const float2*)(ent + (size_t)cc * CH))[lane];
  float* o = out + (size_t)rr * CH + 2 * lane;
  unsafeAtomicAdd(o,     v * ev.x);
  unsafeAtomicAdd(o + 1, v * ev.y);
}

extern "C" void kernel_launch(void* const* d_in, const int* in_sizes, int n_in,
                              void* d_out, int out_size, void* d_ws, size_t ws_size,
                              hipStream_t stream) {
  const float* entity_emb = (const float*)d_in[0];
  // d_in[1] (user_emb) is unused by the reference output
  const int*   edge_index = (const int*)d_in[2];   // [2, E] flat: head row then tail row
  const int*   edge_type  = (const int*)d_in[3];
  const int*   irows      = (const int*)d_in[4];
  const int*   icols      = (const int*)d_in[5];
  const float* ivals      = (const float*)d_in[6];
  const float* rwm        = (const float*)d_in[7];
  const float* relw       = (const float*)d_in[8];

  const int nedges = in_sizes[3];
  const int nnz    = in_sizes[4];

  float* out_ent = (float*)d_out;                      // [N_ENT, CH]
  float* out_usr = out_ent + (size_t)N_ENT * CH;       // [N_USR, CH]

  float* ent    = (float*)d_ws;                        // [N_ENT, CH]
  float* counts = ent + (size_t)N_ENT * CH;            // [N_ENT]

  const int nEntElems = N_ENT * CH;
  const int nTiles    = (RW + 15) / 16;                // 163 M tiles

  zero_kernel<<<(out_size + 255) / 256, 256, 0, stream>>>((float*)d_out, out_size);
  zero_kernel<<<(N_ENT + 255) / 256, 256, 0, stream>>>(counts, N_ENT);
  build_ent_kernel<<<(nEntElems / 4 + 255) / 256, 256, 0, stream>>>(
      (const float4*)entity_emb, (float4*)ent, nEntElems / 4);
  region_gemm_kernel<<<(nTiles + 7) / 8, 256, 0, stream>>>(rwm, entity_emb, ent);
  kg_agg_kernel<<<(nedges + 7) / 8, 256, 0, stream>>>(edge_index, edge_index + nedges,
                                                      edge_type, ent, relw,
                                                      out_ent, counts, nedges);
  finalize_kernel<<<(nEntElems + 255) / 256, 256, 0, stream>>>(out_ent, counts, nEntElems);
  user_agg_kernel<<<(nnz + 7) / 8, 256, 0, stream>>>(irows, icols, ivals, ent, out_usr, nnz);

  (void)out_usr; (void)n_in; (void)ws_size;
}